// MVDRBeamformer_63608465654033
// MI455X (gfx1250) — compile-verified
//
#include <hip/hip_runtime.h>
#include <hip/hip_bf16.h>

#define B_ 8
#define C_ 8
#define F_ 257
#define T_ 1500
#define EPS_ 1e-8f
#define PSD_EPS_ 1e-5f

typedef __attribute__((ext_vector_type(2))) float v2f;
typedef __attribute__((ext_vector_type(8))) float v8f;

// ---------------- Phase 1: per-(b,t) max over F -> reciprocal for mask_norm ----
__global__ void mask_recip_kernel(const float* __restrict__ sm,
                                  const float* __restrict__ nm,
                                  float* __restrict__ recip) {
  int idx = blockIdx.x * blockDim.x + threadIdx.x;
  if (idx >= B_ * T_) return;
  int b = idx / T_;
  int t = idx - b * T_;
  const float* sp = sm + (size_t)b * F_ * T_ + t;
  const float* np = nm + (size_t)b * F_ * T_ + t;
  float ms = 0.f, mn = 0.f;
  for (int f = 0; f < F_; ++f) {
    ms = fmaxf(ms, fabsf(sp[(size_t)f * T_]));
    mn = fmaxf(mn, fabsf(np[(size_t)f * T_]));
  }
  recip[idx] = 1.0f / (ms + EPS_);
  recip[B_ * T_ + idx] = 1.0f / (mn + EPS_);
}

// ---------------- Phase 2: WMMA covariance accumulation ------------------------
// One wave per (b, f, chunk). Astack = [Xr(8) ; Xi(8)] is 16 x T.
// P = Astack * diag(m) * Astack^T via V_WMMA_F32_16X16X4_F32, K=4 per step.
// A lane layout: lane L holds row L%16 at K = {2*(L/16), 2*(L/16)+1}.
// B lane layout mirrors it with N = L%16, so B = A * m(t_of_lane_slot).
__global__ __launch_bounds__(32) void psd_wmma_kernel(
    const float* __restrict__ sm, const float* __restrict__ nm,
    const float* __restrict__ cr, const float* __restrict__ ci,
    const float* __restrict__ recip,
    float* __restrict__ Ppart, float* __restrict__ Dpart,
    int nChunks, int chunkLen) {
  int id = blockIdx.x;
  int b = id / (F_ * nChunks);
  int rem = id - b * (F_ * nChunks);
  int f = rem / nChunks;
  int ch = rem - f * nChunks;
  int t0 = ch * chunkLen;

  int lane = threadIdx.x;
  int row  = lane & 15;      // matrix row (A) == column (B)
  int half = lane >> 4;      // 0: K=0,1   1: K=2,3
  int c    = row & 7;

  const float* xb  = ((row < 8) ? cr : ci) +
                     (((size_t)(b * 8 + c) * F_ + f) * T_) + t0 + 2 * half;
  const float* smp = sm + ((size_t)(b * F_ + f) * T_) + t0 + 2 * half;
  const float* nmp = nm + ((size_t)(b * F_ + f) * T_) + t0 + 2 * half;
  const float* rsp = recip + (size_t)b * T_ + t0 + 2 * half;
  const float* rnp = recip + (size_t)B_ * T_ + (size_t)b * T_ + t0 + 2 * half;

  v8f accS = {0.f, 0.f, 0.f, 0.f, 0.f, 0.f, 0.f, 0.f};
  v8f accN = {0.f, 0.f, 0.f, 0.f, 0.f, 0.f, 0.f, 0.f};
  float denS = 0.f, denN = 0.f;

  int iters = chunkLen >> 2;           // chunkLen divisible by 4
  for (int it = 0; it < iters; ++it) {
    v2f a   = *(const v2f*)xb;
    v2f mrs = *(const v2f*)smp;
    v2f rcs = *(const v2f*)rsp;
    v2f mrn = *(const v2f*)nmp;
    v2f rcn = *(const v2f*)rnp;
    v2f msv = mrs * rcs;               // normalized speech mask at this lane's t
    v2f mnv = mrn * rcn;               // normalized noise  mask
    v2f bs = a * msv;
    v2f bn = a * mnv;
    accS = __builtin_amdgcn_wmma_f32_16x16x4_f32(false, a, false, bs,
                                                 (short)0, accS, false, false);
    accN = __builtin_amdgcn_wmma_f32_16x16x4_f32(false, a, false, bn,
                                                 (short)0, accN, false, false);
    denS += msv.x + msv.y;
    denN += mnv.x + mnv.y;
    xb += 4; smp += 4; nmp += 4; rsp += 4; rnp += 4;
  }

  // lanes 0-15 saw t%4 in {0,1}; lanes 16-31 saw {2,3} -> den = lane0 + lane16
  float dS = __shfl(denS, 0, 32) + __shfl(denS, 16, 32);
  float dN = __shfl(denN, 0, 32) + __shfl(denN, 16, 32);

  // D layout: lane L vgpr j holds P[j + 8*half][L%16]
  float* outP = Ppart + (size_t)id * 512;
#pragma unroll
  for (int j = 0; j < 8; ++j) {
    outP[(j + 8 * half) * 16 + row]       = accS[j];
    outP[256 + (j + 8 * half) * 16 + row] = accN[j];
  }
  if (lane == 0) {
    Dpart[(size_t)id * 2 + 0] = dS;
    Dpart[(size_t)id * 2 + 1] = dN;
  }
}

// ---------------- Phase 3: per-(b,f) 8x8 complex solve + MVDR weights ----------
__global__ void mvdr_weights_kernel(const float* __restrict__ Ppart,
                                    const float* __restrict__ Dpart,
                                    float* __restrict__ wvec, int nChunks) {
  int bf = blockIdx.x * blockDim.x + threadIdx.x;
  if (bf >= B_ * F_) return;

  float Ps[256], Pn[256];
  for (int i = 0; i < 256; ++i) { Ps[i] = 0.f; Pn[i] = 0.f; }
  float dS = 0.f, dN = 0.f;
  for (int chv = 0; chv < nChunks; ++chv) {
    size_t idp = (size_t)bf * nChunks + chv;
    const float* p = Ppart + idp * 512;
    for (int i = 0; i < 256; ++i) { Ps[i] += p[i]; Pn[i] += p[256 + i]; }
    dS += Dpart[idp * 2 + 0];
    dN += Dpart[idp * 2 + 1];
  }
  dS = fmaxf(dS, PSD_EPS_);
  dN = fmaxf(dN, PSD_EPS_);
  float invS = 1.f / dS, invN = 1.f / dN;

  // Augmented complex system [N | S]; nom_re = P[c,d]+P[c+8,d+8],
  // nom_im = P[c+8,d]-P[c,d+8].
  float MR[8][16], MI[8][16];
  float csR[8], csI[8];                 // column sums of S (for inv + i*eps term)
  for (int d = 0; d < 8; ++d) { csR[d] = 0.f; csI[d] = 0.f; }
  for (int r = 0; r < 8; ++r) {
    for (int d = 0; d < 8; ++d) {
      float nr = (Pn[r * 16 + d] + Pn[(r + 8) * 16 + (d + 8)]) * invN;
      float ni = (Pn[(r + 8) * 16 + d] - Pn[r * 16 + (d + 8)]) * invN + PSD_EPS_ + EPS_;
      if (r == d) nr += EPS_;           // + eye*eps before inversion
      float sr = (Ps[r * 16 + d] + Ps[(r + 8) * 16 + (d + 8)]) * invS;
      float si = (Ps[(r + 8) * 16 + d] - Ps[r * 16 + (d + 8)]) * invS + PSD_EPS_ + EPS_;
      MR[r][d] = nr;      MI[r][d] = ni;
      MR[r][8 + d] = sr;  MI[r][8 + d] = si;
      csR[d] += sr;       csI[d] += si;
    }
  }

  // Gauss-Jordan with partial pivoting -> cols 8..15 become X = N^-1 * S
  for (int k = 0; k < 8; ++k) {
    int p = k;
    float best = MR[k][k] * MR[k][k] + MI[k][k] * MI[k][k];
    for (int r = k + 1; r < 8; ++r) {
      float v = MR[r][k] * MR[r][k] + MI[r][k] * MI[r][k];
      if (v > best) { best = v; p = r; }
    }
    if (p != k) {
      for (int j = 0; j < 16; ++j) {
        float tr = MR[k][j]; MR[k][j] = MR[p][j]; MR[p][j] = tr;
        float ti = MI[k][j]; MI[k][j] = MI[p][j]; MI[p][j] = ti;
      }
    }
    float pr = MR[k][k], pi = MI[k][k];
    float dinv = 1.0f / (pr * pr + pi * pi + 1e-30f);
    float qr = pr * dinv, qi = -pi * dinv;      // 1/pivot
    for (int j = 0; j < 16; ++j) {
      float xr = MR[k][j], xi = MI[k][j];
      MR[k][j] = xr * qr - xi * qi;
      MI[k][j] = xr * qi + xi * qr;
    }
    for (int r = 0; r < 8; ++r) {
      if (r == k) continue;
      float fr = MR[r][k], fi = MI[r][k];
      for (int j = 0; j < 16; ++j) {
        float xr = MR[k][j], xi = MI[k][j];
        MR[r][j] -= fr * xr - fi * xi;
        MI[r][j] -= fr * xi + fi * xr;
      }
    }
  }

  // A = X + i*eps*(ones @ S)  (row-independent column sums)
  float trR = EPS_, trI = 0.f;
  for (int k = 0; k < 8; ++k) {
    trR += MR[k][8 + k] - EPS_ * csI[k];
    trI += MI[k][8 + k] + EPS_ * csR[k];
  }
  float den2 = trR * trR + trI * trI;
  float itR = trR / den2, itI = -trI / den2;    // 1/tr

  size_t wb = (size_t)bf * 16;
  for (int r = 0; r < 8; ++r) {
    float ur = MR[r][8] - EPS_ * csI[0];        // u = A[:,0]
    float ui = MI[r][8] + EPS_ * csR[0];
    wvec[wb + r]     = ur * itR - ui * itI;     // weight = u / tr
    wvec[wb + 8 + r] = ur * itI + ui * itR;
  }
}

// ---------------- Phase 4: out[b,f,t] = sum_c conj(w) * cm ---------------------
__global__ void beamform_kernel(const float* __restrict__ cr,
                                const float* __restrict__ ci,
                                const float* __restrict__ wvec,
                                float* __restrict__ out) {
  size_t idx = (size_t)blockIdx.x * blockDim.x + threadIdx.x;
  const size_t total = (size_t)B_ * F_ * T_;
  if (idx >= total) return;
  int t = (int)(idx % T_);
  int bf = (int)(idx / T_);
  int b = bf / F_;
  int f = bf - b * F_;

  const float* w = wvec + (size_t)bf * 16;
  float orl = 0.f, oim = 0.f;
#pragma unroll
  for (int cc = 0; cc < 8; ++cc) {
    float wr = w[cc], wi = w[8 + cc];
    size_t off = (((size_t)(b * 8 + cc) * F_) + f) * T_ + t;
    float xr = cr[off], xi = ci[off];
    orl += wr * xr + wi * xi;          // conj(w) * x
    oim += wr * xi - wi * xr;
  }
  out[idx] = orl;
  out[total + idx] = oim;
}

// ---------------- Launch -------------------------------------------------------
extern "C" void kernel_launch(void* const* d_in, const int* in_sizes, int n_in,
                              void* d_out, int out_size, void* d_ws, size_t ws_size,
                              hipStream_t stream) {
  (void)in_sizes; (void)n_in; (void)out_size;
  const float* sm = (const float*)d_in[0];
  const float* nm = (const float*)d_in[1];
  const float* cr = (const float*)d_in[2];
  const float* ci = (const float*)d_in[3];
  float* out = (float*)d_out;
  float* ws  = (float*)d_ws;

  // ws need (floats): recip 2*B*T  +  P/den partials B*F*nc*514  +  weights B*F*16
  auto need = [](int nc) -> size_t {
    return ((size_t)2 * B_ * T_ + (size_t)B_ * F_ * nc * 514 +
            (size_t)B_ * F_ * 16) * sizeof(float);
  };
  int nChunks = (ws_size >= need(3)) ? 3 : 1;   // 500 and 1500 both % 4 == 0
  int chunkLen = T_ / nChunks;

  float* recip = ws;                                        // 2*B*T
  float* Ppart = recip + (size_t)2 * B_ * T_;               // B*F*nc*512
  float* Dpart = Ppart + (size_t)B_ * F_ * nChunks * 512;   // B*F*nc*2
  float* wvec  = Dpart + (size_t)B_ * F_ * nChunks * 2;     // B*F*16

  mask_recip_kernel<<<(B_ * T_ + 255) / 256, 256, 0, stream>>>(sm, nm, recip);
  psd_wmma_kernel<<<B_ * F_ * nChunks, 32, 0, stream>>>(
      sm, nm, cr, ci, recip, Ppart, Dpart, nChunks, chunkLen);
  mvdr_weights_kernel<<<(B_ * F_ + 63) / 64, 64, 0, stream>>>(Ppart, Dpart, wvec,
                                                              nChunks);
  size_t total = (size_t)B_ * F_ * T_;
  beamform_kernel<<<(unsigned)((total + 255) / 256), 256, 0, stream>>>(cr, ci, wvec,
                                                                       out);
}